// GARCHLSTM_1726576853588
// MI455X (gfx1250) — compile-verified
//
#include <hip/hip_runtime.h>

typedef __attribute__((ext_vector_type(2))) float v2f;
typedef __attribute__((ext_vector_type(4))) float v4f;
typedef __attribute__((ext_vector_type(8))) float v8f;

// Fast transcendental paths: v_exp_f32 + v_rcp_f32 (no IEEE div expansion).
__device__ __forceinline__ float frcp_fast(float x) { return __builtin_amdgcn_rcpf(x); }
__device__ __forceinline__ float fsig(float x)  { return frcp_fast(1.0f + __expf(-x)); }
__device__ __forceinline__ float ftanh(float x) { return 1.0f - 2.0f * frcp_fast(1.0f + __expf(2.0f * x)); }

// One wave processes 16 batch rows per iteration.
// Tile T[h=0..15][b=0..15] of tanh(c_t) lives in WMMA f32 16x16 C/D layout:
//   lane l: column n = l%16 (batch), rows m = (l/16)*8 + v (h), v = vgpr 0..7.
// Column sums (sum over h) via 4x V_WMMA_F32_16X16X4_F32 with all-ones B.
__global__ void __launch_bounds__(256, 2) garchlstm_wmma(
    const float* __restrict__ eps,
    const float* __restrict__ sigma2,
    const float* __restrict__ c_prev,
    const float* __restrict__ Wf_w, const float* __restrict__ Wf_b,
    const float* __restrict__ Uf_w, const float* __restrict__ Uf_b,
    const float* __restrict__ Wi_w, const float* __restrict__ Wi_b,
    const float* __restrict__ Ui_w, const float* __restrict__ Ui_b,
    const float* __restrict__ Wc_w, const float* __restrict__ Wc_b,
    const float* __restrict__ Uc_w, const float* __restrict__ Uc_b,
    const float* __restrict__ b_f,  const float* __restrict__ b_i,
    const float* __restrict__ b_c,  const float* __restrict__ wmix,
    const float* __restrict__ garch_w, const float* __restrict__ garch_b,
    float* __restrict__ out_sig,   // [B]
    float* __restrict__ out_c,     // [B,16]
    int n_groups)                  // B / 16
{
    const int lane   = threadIdx.x & 31;
    const int wgid   = (blockIdx.x * blockDim.x + threadIdx.x) >> 5;
    const int nwaves = (gridDim.x * blockDim.x) >> 5;
    const int bcol   = lane & 15;          // batch column within the 16-row group
    const int hbase  = (lane >> 4) * 8;    // this lane covers h = hbase .. hbase+7

    // ---- loop-invariant scalars / per-lane weight slices (registers) ----
    const float bf = b_f[0], bi = b_i[0], bc = b_c[0];
    const float wm = wmix[0];
    const float gw0 = garch_w[0], gw1 = garch_w[1], gw2 = garch_w[2], gw3 = garch_w[3];
    const float gb  = garch_b[0];

    float wf[8], uf[8], cbf[8];
    float wi[8], ui[8], cbi[8];
    float wc[8], uc[8], cbc[8];
#pragma unroll
    for (int j = 0; j < 8; ++j) {
        const int h = hbase + j;
        wf[j] = Wf_w[h]; uf[j] = Uf_w[h]; cbf[j] = Wf_b[h] + Uf_b[h] + bf;
        wi[j] = Wi_w[h]; ui[j] = Ui_w[h]; cbi[j] = Wi_b[h] + Ui_b[h] + bi;
        wc[j] = Wc_w[h]; uc[j] = Uc_w[h]; cbc[j] = Wc_b[h] + Uc_b[h] + bc;
    }

    const v2f ones = {1.0f, 1.0f};
    const bool valid = (lane < 8) || (lane >= 24);       // these lanes own sigma2_t[b]
    const int  comp  = (lane < 8) ? lane : (lane - 24);  // vgpr holding own column sum

    for (int g = wgid; g < n_groups; g += nwaves) {
        const int b = g * 16 + bcol;

        // prefetch next tile this lane will touch (gfx1250 global_prefetch_b8)
        if (g + nwaves < n_groups) {
            __builtin_prefetch(c_prev + ((size_t)(g + nwaves) * 16 + bcol) * 16 + hbase, 0, 0);
        }

        const float e = eps[b];
        const float s = sigma2[b];

        const v4f* cpv = (const v4f*)(c_prev + (size_t)b * 16 + hbase);
        const v4f cp0 = cpv[0], cp1 = cpv[1];
        float cp[8];
#pragma unroll
        for (int j = 0; j < 4; ++j) { cp[j] = cp0[j]; cp[4 + j] = cp1[j]; }

        float ct[8], tm[8];
#pragma unroll
        for (int j = 0; j < 8; ++j) {
            const float ft = fsig(fmaf(e, wf[j], fmaf(s, uf[j], cbf[j])));
            const float it = fsig(fmaf(e, wi[j], fmaf(s, ui[j], cbi[j])));
            const float ch = fsig(fmaf(e, wc[j], fmaf(s, uc[j], cbc[j])));
            const float c  = fmaf(ft, cp[j], it * ch);
            ct[j] = c;
            tm[j] = ftanh(c);
        }

        // ---- column sums of T via 4 chained f32 WMMAs (A = T vgpr-pairs, B = ones) ----
        v8f acc = {};
        {
            const v2f a0 = {tm[0], tm[1]};
            const v2f a1 = {tm[2], tm[3]};
            const v2f a2 = {tm[4], tm[5]};
            const v2f a3 = {tm[6], tm[7]};
            acc = __builtin_amdgcn_wmma_f32_16x16x4_f32(false, a0, false, ones, (short)0, acc, false, false);
            acc = __builtin_amdgcn_wmma_f32_16x16x4_f32(false, a1, false, ones, (short)0, acc, false, false);
            acc = __builtin_amdgcn_wmma_f32_16x16x4_f32(false, a2, false, ones, (short)0, acc, false, false);
            acc = __builtin_amdgcn_wmma_f32_16x16x4_f32(false, a3, false, ones, (short)0, acc, false, false);
        }

        // store c_t: two fully-coalesced b128 stores per lane (wave covers contiguous 1KB)
        v4f c0, c1;
#pragma unroll
        for (int j = 0; j < 4; ++j) { c0[j] = ct[j]; c1[j] = ct[4 + j]; }
        v4f* co = (v4f*)(out_c + (size_t)b * 16 + hbase);
        co[0] = c0;
        co[1] = c1;

        // extract own column sum (no cross-lane ops needed)
        float sum = acc[0];
#pragma unroll
        for (int k = 1; k < 8; ++k) sum = (comp == k) ? acc[k] : sum;

        if (valid) {
            const float eps2 = e * e;
            const float asym = (e < 0.0f) ? eps2 : 0.0f;
            const float o_t  = fmaf(gw0, 1.0f, fmaf(gw1, eps2, fmaf(gw2, asym, fmaf(gw3, s, gb))));
            out_sig[b] = o_t * fmaf(wm, sum * 0.0625f, 1.0f);
        }
    }
}

extern "C" void kernel_launch(void* const* d_in, const int* in_sizes, int n_in,
                              void* d_out, int out_size, void* d_ws, size_t ws_size,
                              hipStream_t stream) {
    const float* eps     = (const float*)d_in[0];
    const float* sigma2  = (const float*)d_in[1];
    const float* c_prev  = (const float*)d_in[2];
    const float* Wf_w = (const float*)d_in[3];
    const float* Wf_b = (const float*)d_in[4];
    const float* Uf_w = (const float*)d_in[5];
    const float* Uf_b = (const float*)d_in[6];
    const float* Wi_w = (const float*)d_in[7];
    const float* Wi_b = (const float*)d_in[8];
    const float* Ui_w = (const float*)d_in[9];
    const float* Ui_b = (const float*)d_in[10];
    const float* Wc_w = (const float*)d_in[11];
    const float* Wc_b = (const float*)d_in[12];
    const float* Uc_w = (const float*)d_in[13];
    const float* Uc_b = (const float*)d_in[14];
    const float* b_f  = (const float*)d_in[15];
    const float* b_i  = (const float*)d_in[16];
    const float* b_c  = (const float*)d_in[17];
    const float* wmix = (const float*)d_in[18];
    const float* g_w  = (const float*)d_in[19];
    const float* g_b  = (const float*)d_in[20];

    const int n = in_sizes[0];          // B
    const int n_groups = n / 16;        // B = 2^21 -> divisible

    float* out_sig = (float*)d_out;     // outputs concatenated: sigma2_t [B], c_t [B,16]
    float* out_c   = out_sig + n;

    const int threads = 256;
    int blocks = 2048;
    const int max_waves = n_groups;     // never need more waves than groups
    if (blocks * (threads / 32) > max_waves) {
        blocks = (max_waves * 32 + threads - 1) / threads;
        if (blocks < 1) blocks = 1;
    }

    garchlstm_wmma<<<blocks, threads, 0, stream>>>(
        eps, sigma2, c_prev,
        Wf_w, Wf_b, Uf_w, Uf_b,
        Wi_w, Wi_b, Ui_w, Ui_b,
        Wc_w, Wc_b, Uc_w, Uc_b,
        b_f, b_i, b_c, wmix, g_w, g_b,
        out_sig, out_c, n_groups);
}